// NNKNN_19404662243695
// MI455X (gfx1250) — compile-verified
//
#include <hip/hip_runtime.h>
#include <math.h>

typedef float v2f __attribute__((ext_vector_type(2)));
typedef float v4f __attribute__((ext_vector_type(4)));
typedef float v8f __attribute__((ext_vector_type(8)));

#define BB 256
#define NN 4096
#define DD 128
#define CC 32

__device__ __forceinline__ float softplus_f(float x) {
    // numerically safe softplus
    return (x > 20.0f) ? x : log1pf(__expf(x));
}

// ---------------------------------------------------------------------------
// Kernel 1: fused delta + case activations.
// One wave32 per (b, n) pair; lane holds 4 consecutive D-elements (float4).
// delta is a 512 MB write-once stream -> 128-bit non-temporal stores.
// ---------------------------------------------------------------------------
__global__ __launch_bounds__(256) void delta_act_kernel(
    const float* __restrict__ queries,   // [B, D]
    const float* __restrict__ cases,     // [N, D]
    const float* __restrict__ fw,        // [N, D]
    const float* __restrict__ dw,        // [N, D]
    const float* __restrict__ bias,      // [N]
    float* __restrict__ delta_out,       // [B, N, D]
    float* __restrict__ act_out)         // [B, N]
{
    const int wave = threadIdx.x >> 5;
    const int lane = threadIdx.x & 31;
    const long p   = (long)blockIdx.x * 8 + wave;   // pair index in [0, B*N)
    const int b    = (int)(p >> 12);                // p / N   (N = 4096)
    const int n    = (int)(p & (NN - 1));           // p % N
    const int d0   = lane * 4;

    const v4f qv = *(const v4f*)(queries + (long)b * DD + d0);
    const v4f cv = *(const v4f*)(cases   + (long)n * DD + d0);
    const v4f fv = *(const v4f*)(fw      + (long)n * DD + d0);
    const v4f wv = *(const v4f*)(dw      + (long)n * DD + d0);

    v4f sp;
    sp.x = softplus_f(fv.x); sp.y = softplus_f(fv.y);
    sp.z = softplus_f(fv.z); sp.w = softplus_f(fv.w);

    const v4f dif = qv - cv;
    const v4f dl  = dif * dif * sp;                 // delta elements

    // streaming NT store of delta (lane-contiguous 512B per wave)
    __builtin_nontemporal_store(dl, (v4f*)(delta_out + (p << 7) + d0));

    // case_act partial: sum_d delta * (-|dw|)
    float partial = dl.x * (-fabsf(wv.x)) + dl.y * (-fabsf(wv.y)) +
                    dl.z * (-fabsf(wv.z)) + dl.w * (-fabsf(wv.w));

    // wave32 butterfly reduction
    #pragma unroll
    for (int off = 16; off > 0; off >>= 1)
        partial += __shfl_xor(partial, off, 32);

    if (lane == 0) {
        const float s = partial + bias[n];
        act_out[(long)b * NN + n] = 1.0f / (1.0f + __expf(-s));
    }
}

// ---------------------------------------------------------------------------
// Kernel 2: per-row activation sum -> reciprocal denominator in workspace.
// ---------------------------------------------------------------------------
__global__ __launch_bounds__(256) void rowsum_kernel(
    const float* __restrict__ act,   // [B, N]
    float* __restrict__ rinv)        // [B]
{
    __shared__ float sm[256];
    const int b = blockIdx.x;
    float s = 0.0f;
    for (int i = threadIdx.x; i < NN; i += 256)
        s += act[(long)b * NN + i];
    sm[threadIdx.x] = s;
    __syncthreads();
    #pragma unroll
    for (int st = 128; st > 0; st >>= 1) {
        if (threadIdx.x < st) sm[threadIdx.x] += sm[threadIdx.x + st];
        __syncthreads();
    }
    if (threadIdx.x == 0) rinv[b] = 1.0f / (sm[0] + 0.01f);
}

// ---------------------------------------------------------------------------
// Kernel 3: y_hat = diag(rinv) * (act @ targets) via V_WMMA_F32_16X16X4_F32.
// One wave per 16x16 output tile; 16 M-tiles x 2 N-tiles = 32 waves.
// The row normalization is factored out of the K-reduction and applied in
// the epilogue, so the inner loop is pure load -> wmma (no VALU dependency
// in front of each matrix op).
// A layout (ISA 16x4 f32): lanes 0-15 hold K={0,1}, lanes 16-31 hold K={2,3}.
// C/D layout: VGPR j -> M=j (lanes 0-15) / M=8+j (lanes 16-31), N = lane%16.
// ---------------------------------------------------------------------------
__global__ __launch_bounds__(32) void yhat_wmma_kernel(
    const float* __restrict__ act,      // [B, N]
    const float* __restrict__ targets,  // [N, C]
    const float* __restrict__ rinv,     // [B]
    float* __restrict__ y)              // [B, C]
{
    const int lane  = threadIdx.x;
    const int mtile = blockIdx.x >> 1;
    const int ntile = blockIdx.x & 1;
    const int M0 = mtile * 16;
    const int N0 = ntile * 16;
    const int half = lane >> 4;       // 0: K pair {0,1}, 1: K pair {2,3}
    const int l    = lane & 15;

    const float* __restrict__ arow = act + (long)(M0 + l) * NN + 2 * half;
    const float* __restrict__ bcol = targets + (long)(2 * half) * CC + N0 + l;

    v8f acc = {};
    for (int k = 0; k < NN; k += 4) {
        v2f a, b;
        a.x = arow[k];
        a.y = arow[k + 1];
        b.x = bcol[(long)k * CC];
        b.y = bcol[(long)k * CC + CC];
        acc = __builtin_amdgcn_wmma_f32_16x16x4_f32(
            /*neg_a=*/false, a, /*neg_b=*/false, b,
            /*c_mod=*/(short)0, acc, /*reuse_a=*/false, /*reuse_b=*/false);
    }

    // epilogue: apply per-row 1/(rowsum+0.01) and store
    #pragma unroll
    for (int j = 0; j < 8; ++j) {
        const int row = M0 + j + 8 * half;
        y[(long)row * CC + N0 + l] = acc[j] * rinv[row];
    }
}

// ---------------------------------------------------------------------------
extern "C" void kernel_launch(void* const* d_in, const int* in_sizes, int n_in,
                              void* d_out, int out_size, void* d_ws, size_t ws_size,
                              hipStream_t stream) {
    const float* queries = (const float*)d_in[0];   // [B, D]
    const float* cases   = (const float*)d_in[1];   // [N, D]
    const float* targets = (const float*)d_in[2];   // [N, C]
    const float* fw      = (const float*)d_in[3];   // [N, D]
    const float* dw      = (const float*)d_in[4];   // [N, D]
    const float* bias    = (const float*)d_in[5];   // [N]

    // output layout: y_hat [B*C] | activations [B*N] | delta [B*N*D]
    float* y     = (float*)d_out;
    float* act   = y + (size_t)BB * CC;
    float* delta = act + (size_t)BB * NN;
    float* rinv  = (float*)d_ws;                    // [B] scratch

    // 1) fused delta + activations: one wave per (b,n), 8 waves / block
    delta_act_kernel<<<(BB * NN) / 8, 256, 0, stream>>>(
        queries, cases, fw, dw, bias, delta, act);

    // 2) row-wise normalization denominators
    rowsum_kernel<<<BB, 256, 0, stream>>>(act, rinv);

    // 3) WMMA GEMM: y_hat = diag(rinv) * (act @ targets)
    yhat_wmma_kernel<<<(BB / 16) * (CC / 16), 32, 0, stream>>>(
        act, targets, rinv, y);
}